// kWTA_Layer_61546881352183
// MI455X (gfx1250) — compile-verified
//
#include <hip/hip_runtime.h>
#include <stdint.h>

// ---------------------------------------------------------------------------
// kWTA: threshold = k-th largest of N fp32; out[i] = (in[i] < thr) ? 0 : in[i]
//
// 3-level MSB radix select (12 + 12 + 8 bits) on monotonic u32 keys.
// Pass 1 streams 128MB from HBM (stays resident in MI455X's 192MB L2);
// passes 2/3 re-read from L2. Device-side select kernels between passes
// keep everything on-stream / graph-capturable. Final filter pass uses
// 128-bit nontemporal loads/stores.
// ---------------------------------------------------------------------------

typedef float v4f __attribute__((ext_vector_type(4)));

struct SelState {
    uint32_t remaining;   // elements still to account for at current level
    uint32_t prefix;      // high key bits fixed so far
    uint32_t kthKey;      // final 32-bit key of k-th largest
    float    threshold;   // final float threshold
};

__device__ __forceinline__ uint32_t f2key(float f) {
    uint32_t u = __float_as_uint(f);
    // monotonic mapping: larger float -> larger unsigned key
    return (u & 0x80000000u) ? ~u : (u | 0x80000000u);
}

// --------------------------- init ------------------------------------------
__global__ void initKernel(uint32_t* __restrict__ hists, int total,
                           SelState* __restrict__ st,
                           const int* __restrict__ kptr) {
    int i = blockIdx.x * blockDim.x + threadIdx.x;
    if (i < total) hists[i] = 0u;
    if (i == 0) {
        st->remaining = (uint32_t)(*kptr);
        st->prefix    = 0u;
        st->kthKey    = 0u;
        st->threshold = 0.0f;
    }
}

// --------------------------- histogram -------------------------------------
// LEVEL 0: digit = key[31:20]                (4096 bins, no predicate)
// LEVEL 1: digit = key[19:8],  pred key[31:20]==prefix   (4096 bins)
// LEVEL 2: digit = key[7:0],   pred key[31:8] ==prefix   (256 bins)
template <int LEVEL>
__global__ void histKernel(const v4f* __restrict__ in, int nvec, int n,
                           uint32_t* __restrict__ ghist,
                           const SelState* __restrict__ st) {
    constexpr int NB = (LEVEL == 2) ? 256 : 4096;
    __shared__ uint32_t h[NB];
    for (int i = threadIdx.x; i < NB; i += blockDim.x) h[i] = 0u;
    const uint32_t pfx = (LEVEL > 0) ? st->prefix : 0u;  // scalar broadcast
    __syncthreads();

    const int stride = gridDim.x * blockDim.x;
    const int gtid   = blockIdx.x * blockDim.x + threadIdx.x;

    for (int i = gtid; i < nvec; i += stride) {
        v4f v = in[i];  // global_load_b128; RT hint keeps data L2-resident
        #pragma unroll
        for (int c = 0; c < 4; ++c) {
            uint32_t key = f2key(v[c]);
            if (LEVEL == 0) {
                atomicAdd(&h[key >> 20], 1u);
            } else if (LEVEL == 1) {
                if ((key >> 20) == pfx) atomicAdd(&h[(key >> 8) & 0xFFFu], 1u);
            } else {
                if ((key >> 8) == pfx) atomicAdd(&h[key & 0xFFu], 1u);
            }
        }
    }
    // scalar tail (N is a multiple of 4 here, but stay general)
    int tailBase = nvec * 4;
    if (tailBase + gtid < n) {
        uint32_t key = f2key(((const float*)in)[tailBase + gtid]);
        if (LEVEL == 0) {
            atomicAdd(&h[key >> 20], 1u);
        } else if (LEVEL == 1) {
            if ((key >> 20) == pfx) atomicAdd(&h[(key >> 8) & 0xFFFu], 1u);
        } else {
            if ((key >> 8) == pfx) atomicAdd(&h[key & 0xFFu], 1u);
        }
    }
    __syncthreads();
    // merge privatized histogram; skip empty bins (normal data occupies few)
    for (int i = threadIdx.x; i < NB; i += blockDim.x) {
        uint32_t c = h[i];
        if (c) atomicAdd(&ghist[i], c);
    }
}

// --------------------------- bucket select ---------------------------------
// Single workgroup of 256 threads. Finds largest bin b with
// count(bins above b) < remaining <= count(bins >= b), then updates state.
__global__ void selectKernel(const uint32_t* __restrict__ hist, int nbins,
                             int digitBits, int isLast,
                             SelState* __restrict__ st) {
    __shared__ uint32_t sc[256];
    const int t   = threadIdx.x;
    const int bpt = nbins / 256;  // bins per thread (16 or 1)

    // read state into registers before any write can occur
    const uint32_t rem       = st->remaining;
    const uint32_t oldPrefix = st->prefix;

    // partial = count over my chunk of bins (chunks ordered top-bin-first)
    uint32_t partial = 0;
    for (int j = 0; j < bpt; ++j) {
        int r   = t * bpt + j;        // reversed index: r=0 is top bin
        int bin = nbins - 1 - r;
        partial += hist[bin];
    }
    sc[t] = partial;
    __syncthreads();
    // inclusive Hillis-Steele scan over 256 partials
    for (int off = 1; off < 256; off <<= 1) {
        uint32_t v = (t >= off) ? sc[t - off] : 0u;
        __syncthreads();
        sc[t] += v;
        __syncthreads();
    }
    uint32_t cum = sc[t] - partial;   // count in all bins above my chunk

    for (int j = 0; j < bpt; ++j) {
        int r   = t * bpt + j;
        int bin = nbins - 1 - r;
        uint32_t c = hist[bin];
        if (cum < rem && cum + c >= rem) {   // exactly one (t,j) hits this
            uint32_t newPrefix = (oldPrefix << digitBits) | (uint32_t)bin;
            st->prefix    = newPrefix;
            st->remaining = rem - cum;
            if (isLast) {
                uint32_t key = newPrefix;
                uint32_t u = (key & 0x80000000u) ? (key ^ 0x80000000u) : ~key;
                st->kthKey    = key;
                st->threshold = __uint_as_float(u);
            }
        }
        cum += c;
    }
}

// --------------------------- filter ----------------------------------------
__global__ void filterKernel(const v4f* __restrict__ in, v4f* __restrict__ out,
                             int nvec, int n, const SelState* __restrict__ st) {
    const float thr  = st->threshold;   // scalar broadcast load
    const int stride = gridDim.x * blockDim.x;
    const int gtid   = blockIdx.x * blockDim.x + threadIdx.x;

    for (int i = gtid; i < nvec; i += stride) {
        if (i + stride < nvec)
            __builtin_prefetch(in + i + stride, 0, 0);  // global_prefetch_b8
        // last use of the input: nontemporal load; output never re-read: NT store
        v4f v = __builtin_nontemporal_load(in + i);
        v4f r;
        r.x = (v.x < thr) ? 0.0f : v.x;
        r.y = (v.y < thr) ? 0.0f : v.y;
        r.z = (v.z < thr) ? 0.0f : v.z;
        r.w = (v.w < thr) ? 0.0f : v.w;
        __builtin_nontemporal_store(r, out + i);
    }
    int tailBase = nvec * 4;
    if (tailBase + gtid < n) {
        float x = ((const float*)in)[tailBase + gtid];
        ((float*)out)[tailBase + gtid] = (x < thr) ? 0.0f : x;
    }
}

// --------------------------- launch ----------------------------------------
extern "C" void kernel_launch(void* const* d_in, const int* in_sizes, int n_in,
                              void* d_out, int out_size, void* d_ws,
                              size_t ws_size, hipStream_t stream) {
    (void)n_in; (void)out_size; (void)ws_size;

    const float* x    = (const float*)d_in[0];
    const int*   kptr = (const int*)d_in[1];
    const int n    = in_sizes[0];
    const int nvec = n >> 2;

    uint32_t* hist1 = (uint32_t*)d_ws;          // 4096
    uint32_t* hist2 = hist1 + 4096;             // 4096
    uint32_t* hist3 = hist2 + 4096;             // 256
    SelState* st    = (SelState*)(hist3 + 256);
    const int histTotal = 4096 + 4096 + 256;

    const v4f* inv  = (const v4f*)x;
    v4f*       outv = (v4f*)d_out;

    // 1) zero histograms, load k into state
    initKernel<<<(histTotal + 255) / 256, 256, 0, stream>>>(hist1, histTotal,
                                                            st, kptr);
    // 2) level 0: top 12 bits (streams 128MB from HBM -> resident in 192MB L2)
    histKernel<0><<<2048, 256, 0, stream>>>(inv, nvec, n, hist1, st);
    selectKernel<<<1, 256, 0, stream>>>(hist1, 4096, 12, 0, st);
    // 3) level 1: middle 12 bits (L2 hits)
    histKernel<1><<<2048, 256, 0, stream>>>(inv, nvec, n, hist2, st);
    selectKernel<<<1, 256, 0, stream>>>(hist2, 4096, 12, 0, st);
    // 4) level 2: low 8 bits (L2 hits) -> exact k-th largest value
    histKernel<2><<<2048, 256, 0, stream>>>(inv, nvec, n, hist3, st);
    selectKernel<<<1, 256, 0, stream>>>(hist3, 256, 8, 1, st);
    // 5) filter: NT b128 load/store stream, ~11us HBM floor
    filterKernel<<<4096, 256, 0, stream>>>(inv, outv, nvec, n, st);
}